// GraphConvGRUCell_82669530513657
// MI455X (gfx1250) — compile-verified
//
#include <hip/hip_runtime.h>

// Problem constants from the reference.
#define B_   16
#define N_   10000
#define H_   128
#define IN_  256
#define DEG_ 16
#define E_   (B_ * N_ * DEG_)   // 2,560,000 edges
#define NN_  (B_ * N_)          // 160,000 nodes (NN_ % 256 == 0, NN_/256 = 625)
#define NBLK_ (NN_ / 256)       // 625 scan blocks

typedef float v2f __attribute__((ext_vector_type(2)));
typedef float v8f __attribute__((ext_vector_type(8)));

// ---------------------------------------------------------------------------
// 1) Integer degree counts per node (out-degree from src, in-degree from dst).
// ---------------------------------------------------------------------------
__global__ void k_degrees(const int* __restrict__ src, const int* __restrict__ dst,
                          int* __restrict__ cntO, int* __restrict__ cntI) {
    int e = blockIdx.x * blockDim.x + threadIdx.x;
    if (e < E_) {
        atomicAdd(&cntO[src[e]], 1);
        atomicAdd(&cntI[dst[e]], 1);
    }
}

// ---------------------------------------------------------------------------
// 2) norm_out / norm_in = rsqrt(max(deg, 1))
// ---------------------------------------------------------------------------
__global__ void k_norm(const int* __restrict__ cntO, const int* __restrict__ cntI,
                       float* __restrict__ normO, float* __restrict__ normI) {
    int i = blockIdx.x * blockDim.x + threadIdx.x;
    if (i < NN_) {
        normO[i] = rsqrtf(fmaxf((float)cntO[i], 1.0f));
        normI[i] = rsqrtf(fmaxf((float)cntI[i], 1.0f));
    }
}

// ---------------------------------------------------------------------------
// 3) Exclusive scan of in-degrees -> CSR row starts (3-kernel scan).
// ---------------------------------------------------------------------------
__global__ void k_scan1(const int* __restrict__ cnt, int* __restrict__ rowst,
                        int* __restrict__ bsum) {
    __shared__ int sh[256];
    int tid = threadIdx.x;
    int i = blockIdx.x * 256 + tid;
    int v = (i < NN_) ? cnt[i] : 0;
    sh[tid] = v;
    __syncthreads();
    for (int off = 1; off < 256; off <<= 1) {
        int t = (tid >= off) ? sh[tid - off] : 0;
        __syncthreads();
        sh[tid] += t;
        __syncthreads();
    }
    if (i < NN_) rowst[i] = sh[tid] - v;          // exclusive within block
    if (tid == 255) bsum[blockIdx.x] = sh[255];   // block total
}

__global__ void k_scan2(const int* __restrict__ bsum, int* __restrict__ boff) {
    __shared__ int sh[1024];
    int tid = threadIdx.x;
    int v = (tid < NBLK_) ? bsum[tid] : 0;
    sh[tid] = v;
    __syncthreads();
    for (int off = 1; off < 1024; off <<= 1) {
        int t = (tid >= off) ? sh[tid - off] : 0;
        __syncthreads();
        sh[tid] += t;
        __syncthreads();
    }
    if (tid < NBLK_) boff[tid] = sh[tid] - v;     // exclusive block offsets
}

__global__ void k_scan3(int* __restrict__ rowst, const int* __restrict__ boff,
                        int* __restrict__ cursor) {
    int i = blockIdx.x * 256 + threadIdx.x;
    if (i < NN_) {
        int r = rowst[i] + boff[blockIdx.x];
        rowst[i]  = r;
        cursor[i] = r;
    }
}

// ---------------------------------------------------------------------------
// 4) Bucket fill: for each edge, append its SOURCE node id to dst's bucket.
//    (Each slot in [0, E_) written exactly once; cursors partition the range.)
// ---------------------------------------------------------------------------
__global__ void k_fill(const int* __restrict__ src, const int* __restrict__ dst,
                       int* __restrict__ cursor, int* __restrict__ edgesrc) {
    int e = blockIdx.x * blockDim.x + threadIdx.x;
    if (e < E_) {
        int pos = atomicAdd(&cursor[dst[e]], 1);
        edgesrc[pos] = src[e];
    }
}

// ---------------------------------------------------------------------------
// 5) Tiny GEMMs: xr/xz/xh = x @ W_{r,z,h} + b   ([16,256]@[256,128], negligible)
// ---------------------------------------------------------------------------
__global__ void k_xgemm(const float* __restrict__ x,
                        const float* __restrict__ Wr, const float* __restrict__ br,
                        const float* __restrict__ Wz, const float* __restrict__ bz,
                        const float* __restrict__ Wh, const float* __restrict__ bh,
                        float* __restrict__ xr, float* __restrict__ xz, float* __restrict__ xh) {
    int t = blockIdx.x * blockDim.x + threadIdx.x;   // 3 * 16 * 128 = 6144 threads
    if (t >= 3 * B_ * H_) return;
    int w   = t / (B_ * H_);
    int rem = t % (B_ * H_);
    int b = rem / H_, c = rem % H_;
    const float* W  = (w == 0) ? Wr : (w == 1) ? Wz : Wh;
    const float* bb = (w == 0) ? br : (w == 1) ? bz : bh;
    float*       o  = (w == 0) ? xr : (w == 1) ? xz : xh;
    float s = bb[c];
    for (int k = 0; k < IN_; ++k)
        s += x[b * IN_ + k] * W[k * H_ + c];
    o[rem] = s;
}

// ---------------------------------------------------------------------------
// 6) Fully fused: CSR gather -> LDS A-strip -> fp32 WMMA GEMM (@ W_g + b_g)
//    -> GRU elementwise epilogue -> final output. The h_conv / agg
//    intermediates never touch global memory.
//
//    Block = 256 threads = 8 waves; block owns 16 dst nodes (rows).
//    Gather: wave w accumulates rows w and w+8; lane holds 4 of 128 cols
//    (float4), scaled by norm_out[src] per edge and norm_in[row] at the end.
//    GEMM: wave w owns output cols [16w,16w+16); 32 x v_wmma_f32_16x16x4_f32.
//    ISA layouts (CDNA5 ISA):
//      A 16x4 : lanes 0-15 -> M=lane,    v.x=K0, v.y=K1
//               lanes16-31 -> M=lane-16, v.x=K2, v.y=K3
//      B 4x16 : mirrored (N=lane&15, rows K0/K1 vs K2/K3 per half)
//      C/D    : VGPR i -> M = i (lanes 0-15) / M = i+8 (lanes 16-31), N = lane&15
// ---------------------------------------------------------------------------
__global__ void __launch_bounds__(256)
k_gather_gemm_gru(const int* __restrict__ rowst, const int* __restrict__ cntI,
                  const int* __restrict__ edgesrc,
                  const float* __restrict__ normO, const float* __restrict__ normI,
                  const float* __restrict__ h_prev,
                  const float* __restrict__ Wg, const float* __restrict__ bg,
                  const float* __restrict__ xr, const float* __restrict__ xz,
                  const float* __restrict__ xh, float* __restrict__ out) {
    __shared__ float As[16 * H_];   // 8 KB normalized A-strip, shared by 8 waves

    const int m0   = blockIdx.x * 16;
    const int tid  = threadIdx.x;
    const int wave = tid >> 5;
    const int lane = tid & 31;

    const float4* __restrict__ hp4 = reinterpret_cast<const float4*>(h_prev);

    // ---- CSR gather: each wave produces 2 of the 16 rows ----
    for (int rr = wave; rr < 16; rr += 8) {
        const int node = m0 + rr;
        const int beg  = rowst[node];
        const int end  = beg + cntI[node];
        float4 acc = make_float4(0.f, 0.f, 0.f, 0.f);
        for (int j = beg; j < end; ++j) {
            const int s = edgesrc[j];          // uniform across the wave
            const float no = normO[s];
            const float4 v = hp4[s * (H_ / 4) + lane];
            acc.x += v.x * no;
            acc.y += v.y * no;
            acc.z += v.z * no;
            acc.w += v.w * no;
        }
        const float ni = normI[node];
        acc.x *= ni; acc.y *= ni; acc.z *= ni; acc.w *= ni;
        reinterpret_cast<float4*>(As)[rr * (H_ / 4) + lane] = acc;  // ds_store_b128
    }
    __syncthreads();

    // ---- fp32 WMMA: A(16x128, LDS) x W_g(128x128, cached global) ----
    const int n0   = wave * 16;
    const int half = lane >> 4;     // 0: K pair {0,1}, 1: K pair {2,3}
    const int l15  = lane & 15;

    v8f c = {};
#pragma unroll
    for (int k0 = 0; k0 < H_; k0 += 4) {
        const int ka = k0 + half * 2;
        v2f a, b;
        a.x = As[l15 * H_ + ka];
        a.y = As[l15 * H_ + ka + 1];
        b.x = Wg[ka * H_ + n0 + l15];
        b.y = Wg[(ka + 1) * H_ + n0 + l15];
        c = __builtin_amdgcn_wmma_f32_16x16x4_f32(
                /*neg_a=*/false, a, /*neg_b=*/false, b,
                /*c_mod=*/(short)0, c, /*reuse_a=*/false, /*reuse_b=*/false);
    }

    // ---- GRU epilogue directly on the accumulator tile ----
    const int col = n0 + l15;
    const float bgc = bg[col];
#pragma unroll
    for (int i = 0; i < 8; ++i) {
        int row   = m0 + i + half * 8;      // global node index
        int batch = row / N_;               // block-diagonal batch id
        float hc  = c[i] + bgc;
        float r_t = 1.0f / (1.0f + __expf(-(xr[batch * H_ + col] + hc)));
        float z_t = 1.0f / (1.0f + __expf(-(xz[batch * H_ + col] + hc)));
        float h_t = tanhf(xh[batch * H_ + col] + r_t * hc);
        float hp  = h_prev[row * H_ + col];
        out[row * H_ + col] = (1.0f - z_t) * hp + z_t * h_t;
    }
}

// ---------------------------------------------------------------------------
// Launcher. Workspace layout (ints then floats), ~14.1 MB total:
//   cntO, cntI, rowst, cursor : NN_ ints each
//   bsum, boff                : 1024 ints each
//   edgesrc                   : E_ ints
//   normO, normI              : NN_ floats each
//   xr, xz, xh                : B_*H_ floats each
// ---------------------------------------------------------------------------
extern "C" void kernel_launch(void* const* d_in, const int* in_sizes, int n_in,
                              void* d_out, int out_size, void* d_ws, size_t ws_size,
                              hipStream_t stream) {
    const float* x      = (const float*)d_in[0];
    const float* h_prev = (const float*)d_in[1];
    const int*   src    = (const int*)d_in[2];
    const int*   dst    = (const int*)d_in[3];
    const float* Wr     = (const float*)d_in[4];
    const float* br     = (const float*)d_in[5];
    const float* Wz     = (const float*)d_in[6];
    const float* bz     = (const float*)d_in[7];
    const float* Wh     = (const float*)d_in[8];
    const float* bh     = (const float*)d_in[9];
    const float* Wg     = (const float*)d_in[10];
    const float* bg     = (const float*)d_in[11];
    float* out = (float*)d_out;

    int* wsi     = (int*)d_ws;
    int* cntO    = wsi;
    int* cntI    = cntO + NN_;
    int* rowst   = cntI + NN_;
    int* cursor  = rowst + NN_;
    int* bsum    = cursor + NN_;
    int* boff    = bsum + 1024;
    int* edgesrc = boff + 1024;
    float* normO = (float*)(edgesrc + E_);
    float* normI = normO + NN_;
    float* xr    = normI + NN_;
    float* xz    = xr + B_ * H_;
    float* xh    = xz + B_ * H_;

    // Zero only the degree counters (everything else is fully overwritten).
    hipMemsetAsync(cntO, 0, 2 * (size_t)NN_ * sizeof(int), stream);

    k_degrees<<<(E_ + 255) / 256, 256, 0, stream>>>(src, dst, cntO, cntI);
    k_norm<<<NBLK_, 256, 0, stream>>>(cntO, cntI, normO, normI);
    k_scan1<<<NBLK_, 256, 0, stream>>>(cntI, rowst, bsum);
    k_scan2<<<1, 1024, 0, stream>>>(bsum, boff);
    k_scan3<<<NBLK_, 256, 0, stream>>>(rowst, boff, cursor);
    k_fill<<<(E_ + 255) / 256, 256, 0, stream>>>(src, dst, cursor, edgesrc);
    k_xgemm<<<(3 * B_ * H_ + 255) / 256, 256, 0, stream>>>(x, Wr, br, Wz, bz, Wh, bh,
                                                           xr, xz, xh);
    k_gather_gemm_gru<<<NN_ / 16, 256, 0, stream>>>(rowst, cntI, edgesrc, normO, normI,
                                                    h_prev, Wg, bg, xr, xz, xh, out);
}